// Attention_6519760355548
// MI455X (gfx1250) — compile-verified
//
#include <hip/hip_runtime.h>
#include <math.h>

typedef __attribute__((ext_vector_type(2))) float v2f;
typedef __attribute__((ext_vector_type(8))) float v8f;

#define N_SENT   65536
#define N_BAGS   4096
#define REL_DIM  768
#define FLAT_C   53
#define KTOT     (REL_DIM * 3)   // 2304
#define CHUNK    128

// ---------------------------------------------------------------------------
// Kernel A: per-bag online-softmax attention + weighted aggregation.
// One 256-thread block (8 waves) per bag. Single logical pass over x:
// phase-2 re-reads the current 128-sentence chunk which is L2-hot.
// ---------------------------------------------------------------------------
__global__ __launch_bounds__(256)
void bag_attention_kernel(const float* __restrict__ x,        // [N_SENT, 768]
                          const int*   __restrict__ aq,       // [N_SENT, 3]
                          const int*   __restrict__ scope,    // [N_BAGS+1]
                          const float* __restrict__ attw,     // [95, 768]
                          float* __restrict__ out_stack,      // [3, N_BAGS, 768]
                          float* __restrict__ out_total)      // [N_BAGS, 2304]
{
    const int b    = blockIdx.x;
    const int t    = threadIdx.x;
    const int wave = t >> 5;
    const int lane = t & 31;

    const int start = scope[b];
    const int count = scope[b + 1] - start;

    __shared__ float sE[CHUNK * 3];     // chunk logits, then exp-weights
    __shared__ float sRed[256 * 3];     // block reductions (3 layers)
    __shared__ float sM[3];             // running max
    __shared__ float sS[3];             // running (rescaled) sum
    __shared__ float sR[3];             // rescale factor for this chunk
    __shared__ float sCM[3];            // new max for this chunk

    if (t == 0) {
        sM[0] = sM[1] = sM[2] = -INFINITY;
        sS[0] = sS[1] = sS[2] = 0.0f;
    }

    // acc[layer][j] : this thread owns dims d = t + 256*j, j = 0..2
    float acc[3][3];
#pragma unroll
    for (int l = 0; l < 3; ++l)
#pragma unroll
        for (int j = 0; j < 3; ++j) acc[l][j] = 0.0f;

    for (int base = 0; base < count; base += CHUNK) {
        const int cc = min(CHUNK, count - base);
        __syncthreads();  // prev phase-2 done (sE reuse); first iter: init visible

        // ---- phase 1: logits for this chunk (one wave per sentence) ----
        for (int i = wave; i < cc; i += 8) {
            const int n  = start + base + i;
            const int q0 = aq[n * 3 + 0];
            const int q1 = aq[n * 3 + 1];
            const int q2 = aq[n * 3 + 2];
            const float* xr = x    + (size_t)n  * REL_DIM;
            const float* a0 = attw + (size_t)q0 * REL_DIM;
            const float* a1 = attw + (size_t)q1 * REL_DIM;
            const float* a2 = attw + (size_t)q2 * REL_DIM;
            float p0 = 0.f, p1 = 0.f, p2 = 0.f;
            for (int d = lane; d < REL_DIM; d += 32) {
                const float xv = xr[d];
                p0 = fmaf(xv, a0[d], p0);
                p1 = fmaf(xv, a1[d], p1);
                p2 = fmaf(xv, a2[d], p2);
            }
#pragma unroll
            for (int off = 16; off > 0; off >>= 1) {
                p0 += __shfl_xor(p0, off, 32);
                p1 += __shfl_xor(p1, off, 32);
                p2 += __shfl_xor(p2, off, 32);
            }
            if (lane == 0) {
                sE[i * 3 + 0] = p0;
                sE[i * 3 + 1] = p1;
                sE[i * 3 + 2] = p2;
            }
        }
        __syncthreads();

        // ---- chunk max per layer (block reduction) ----
        float lm0 = -INFINITY, lm1 = -INFINITY, lm2 = -INFINITY;
        for (int i = t; i < cc; i += 256) {
            lm0 = fmaxf(lm0, sE[i * 3 + 0]);
            lm1 = fmaxf(lm1, sE[i * 3 + 1]);
            lm2 = fmaxf(lm2, sE[i * 3 + 2]);
        }
        sRed[t] = lm0; sRed[t + 256] = lm1; sRed[t + 512] = lm2;
        __syncthreads();
        for (int s = 128; s > 0; s >>= 1) {
            if (t < s) {
                sRed[t]       = fmaxf(sRed[t],       sRed[t + s]);
                sRed[t + 256] = fmaxf(sRed[t + 256], sRed[t + 256 + s]);
                sRed[t + 512] = fmaxf(sRed[t + 512], sRed[t + 512 + s]);
            }
            __syncthreads();
        }
        if (t == 0) {
#pragma unroll
            for (int l = 0; l < 3; ++l) {
                const float nm = fmaxf(sM[l], sRed[l * 256]);
                sR[l]  = __expf(sM[l] - nm);   // 0 on first chunk (-inf - finite)
                sM[l]  = nm;
                sCM[l] = nm;
            }
        }
        __syncthreads();

        // ---- exponentiate + chunk sum per layer ----
        float ls0 = 0.f, ls1 = 0.f, ls2 = 0.f;
        const float cm0 = sCM[0], cm1 = sCM[1], cm2 = sCM[2];
        for (int i = t; i < cc; i += 256) {
            const float e0 = __expf(sE[i * 3 + 0] - cm0);
            const float e1 = __expf(sE[i * 3 + 1] - cm1);
            const float e2 = __expf(sE[i * 3 + 2] - cm2);
            sE[i * 3 + 0] = e0; sE[i * 3 + 1] = e1; sE[i * 3 + 2] = e2;
            ls0 += e0; ls1 += e1; ls2 += e2;
        }
        sRed[t] = ls0; sRed[t + 256] = ls1; sRed[t + 512] = ls2;
        __syncthreads();
        for (int s = 128; s > 0; s >>= 1) {
            if (t < s) {
                sRed[t]       += sRed[t + s];
                sRed[t + 256] += sRed[t + 256 + s];
                sRed[t + 512] += sRed[t + 512 + s];
            }
            __syncthreads();
        }
        if (t == 0) {
#pragma unroll
            for (int l = 0; l < 3; ++l)
                sS[l] = sS[l] * sR[l] + sRed[l * 256];
        }

        // ---- rescale running accumulators ----
        const float r0 = sR[0], r1 = sR[1], r2 = sR[2];
#pragma unroll
        for (int j = 0; j < 3; ++j) {
            acc[0][j] *= r0; acc[1][j] *= r1; acc[2][j] *= r2;
        }

        // ---- phase 2: accumulate w * x (chunk x is L2-hot) ----
        for (int i = 0; i < cc; ++i) {
            const float e0 = sE[i * 3 + 0];
            const float e1 = sE[i * 3 + 1];
            const float e2 = sE[i * 3 + 2];
            const float* xr = x + (size_t)(start + base + i) * REL_DIM;
#pragma unroll
            for (int j = 0; j < 3; ++j) {
                const float xv = xr[t + 256 * j];
                acc[0][j] = fmaf(e0, xv, acc[0][j]);
                acc[1][j] = fmaf(e1, xv, acc[1][j]);
                acc[2][j] = fmaf(e2, xv, acc[2][j]);
            }
        }
    }
    __syncthreads();

    const float inv0 = 1.0f / sS[0];
    const float inv1 = 1.0f / sS[1];
    const float inv2 = 1.0f / sS[2];
    float* tot = out_total + (size_t)b * KTOT;
#pragma unroll
    for (int j = 0; j < 3; ++j) {
        const int d = t + 256 * j;
        const float v0 = acc[0][j] * inv0;
        const float v1 = acc[1][j] * inv1;
        const float v2 = acc[2][j] * inv2;
        out_stack[(size_t)0 * N_BAGS * REL_DIM + (size_t)b * REL_DIM + d] = v0;
        out_stack[(size_t)1 * N_BAGS * REL_DIM + (size_t)b * REL_DIM + d] = v1;
        out_stack[(size_t)2 * N_BAGS * REL_DIM + (size_t)b * REL_DIM + d] = v2;
        tot[0 * REL_DIM + d] = v0;
        tot[1 * REL_DIM + d] = v1;
        tot[2 * REL_DIM + d] = v2;
    }
}

// ---------------------------------------------------------------------------
// Kernel B: probs = logits_total[4096,2304] @ relation_weight^T[2304,53] + bias
// via V_WMMA_F32_16X16X4_F32. 8 waves/block, each wave owns one 16x16 tile
// (wave w -> M-tile blockIdx.x*8+w; N-tile blockIdx.y). Columns >= 53 use
// clamped addresses and are never stored; EXEC stays all-1s around the WMMA.
// ---------------------------------------------------------------------------
__global__ __launch_bounds__(256)
void probs_wmma_kernel(const float* __restrict__ A,     // [4096, 2304]
                       const float* __restrict__ W,     // [53, 2304]
                       const float* __restrict__ bias,  // [53]
                       float* __restrict__ out)         // [4096, 53]
{
    const int lane  = threadIdx.x & 31;
    const int wave  = threadIdx.x >> 5;
    const int mTile = blockIdx.x * 8 + wave;
    const int nTile = blockIdx.y;

    const int l15  = lane & 15;
    const int half = lane >> 4;           // 0: lanes 0-15, 1: lanes 16-31

    // A 16x4 f32 layout: lane l15 holds M=l15; VGPR0=K(2*half), VGPR1=K(2*half+1)
    const int row  = mTile * 16 + l15;
    // B 4x16 f32 layout: lane l15 holds N=l15; VGPR0=K(2*half), VGPR1=K(2*half+1)
    const int col  = nTile * 16 + l15;
    const int ccol = (col < FLAT_C) ? col : (FLAT_C - 1);  // clamp addr only

    const float* aPtr = A + (size_t)row  * KTOT + 2 * half;
    const float* wPtr = W + (size_t)ccol * KTOT + 2 * half;

    v8f c = {};
    for (int k = 0; k < KTOT; k += 4) {
        const v2f a  = *(const v2f*)(aPtr + k);   // K = k+2*half, k+2*half+1
        const v2f bv = *(const v2f*)(wPtr + k);   // B[k][n] = W[n][k]
        c = __builtin_amdgcn_wmma_f32_16x16x4_f32(
                /*neg_a=*/false, a, /*neg_b=*/false, bv,
                /*c_mod=*/(short)0, c, /*reuse_a=*/false, /*reuse_b=*/false);
    }

    // D 16x16 f32 layout: VGPR r -> M = r + 8*half, N = l15
    if (col < FLAT_C) {
        const float bb = bias[col];
        const int mBase = mTile * 16 + 8 * half;
#pragma unroll
        for (int r = 0; r < 8; ++r) {
            out[(size_t)(mBase + r) * FLAT_C + col] = c[r] + bb;
        }
    }
}

extern "C" void kernel_launch(void* const* d_in, const int* in_sizes, int n_in,
                              void* d_out, int out_size, void* d_ws, size_t ws_size,
                              hipStream_t stream) {
    (void)in_sizes; (void)n_in; (void)out_size; (void)d_ws; (void)ws_size;

    const float* x     = (const float*)d_in[0];   // [65536, 768]
    const int*   aq    = (const int*)  d_in[1];   // [65536, 3]
    const int*   scope = (const int*)  d_in[2];   // [4097]
    const float* attw  = (const float*)d_in[3];   // [95, 768]
    const float* relw  = (const float*)d_in[4];   // [53, 2304]
    const float* bias  = (const float*)d_in[5];   // [53]

    float* out       = (float*)d_out;
    float* out_stack = out;                                             // [3,4096,768]
    float* out_total = out + (size_t)3 * N_BAGS * REL_DIM;              // [4096,2304]
    float* out_probs = out_total + (size_t)N_BAGS * KTOT;               // [4096,53]

    bag_attention_kernel<<<N_BAGS, 256, 0, stream>>>(
        x, aq, scope, attw, out_stack, out_total);

    dim3 grid(N_BAGS / (16 * 8), (FLAT_C + 15) / 16, 1);                // (32, 4)
    probs_wmma_kernel<<<grid, 256, 0, stream>>>(
        out_total, relw, bias, out_probs);
}